// RandomizedHadamardTransform_25116968747683
// MI455X (gfx1250) — compile-verified
//
#include <hip/hip_runtime.h>

typedef __attribute__((ext_vector_type(2))) float v2f;
typedef __attribute__((ext_vector_type(4))) float v4f;
typedef __attribute__((ext_vector_type(8))) float v8f;

#define WAVES_PER_BLOCK 4
#define ROW_D 4096

// Per row (4096 f32) viewed as T[i][j][k], i,j,k in [0,16):
//   FWHT_4096 = (H16 along i) o (H16 along j) o (H16 along k),  scale 1/64.
// j/k axes via chained V_WMMA_F32_16X16X4_F32; i axis via register butterflies.
__global__ __launch_bounds__(128) void rht4096_kernel(const float* __restrict__ x,
                                                      const float* __restrict__ signs,
                                                      float* __restrict__ out) {
    __shared__ __align__(16) float lds[WAVES_PER_BLOCK * ROW_D];

    const int tid  = threadIdx.x;
    const int wave = tid >> 5;
    const int lane = tid & 31;
    const int half = lane >> 4;     // which 16-lane half of the wave
    const int m    = lane & 15;     // row (A layout) / col (B,C,D layout) index

    float* wbuf = &lds[wave * ROW_D];
    const long long row = (long long)blockIdx.x * WAVES_PER_BLOCK + wave;
    const float* xrow = x   + row * ROW_D;
    float*       orow = out + row * ROW_D;

    // ---- Stage 0: stream row in, apply signs, stage row-major in LDS ----
    // d = i*256 + j*16 + k ; lane grabs 4 contiguous floats per chunk.
#pragma unroll 4
    for (int it = 0; it < 32; ++it) {
        const int idx = it * 128 + lane * 4;
        v4f xv = __builtin_nontemporal_load((const v4f*)(xrow + idx));
        v4f sv = *(const v4f*)(signs + idx);          // 16KB, stays cached
        *(v4f*)(wbuf + idx) = xv * sv;
    }

    // ---- H16 chunk constants (A-layout == B-layout since H16 symmetric) ----
    // hc[kc] = { H[4kc+2*half][m], H[4kc+2*half+1][m] },  H[a][b] = (-1)^popc(a&b)
    v2f hc[4];
#pragma unroll
    for (int kc = 0; kc < 4; ++kc) {
        const int r0 = 4 * kc + 2 * half;
        hc[kc].x = 1.0f - 2.0f * (float)(__builtin_popcount(r0 & m) & 1);
        hc[kc].y = 1.0f - 2.0f * (float)(__builtin_popcount((r0 + 1) & m) & 1);
    }

    // ---- Stages 1+2: per 16x16 tile,  V_i = H16 * (T_i * H16)  via WMMA ----
    for (int i = 0; i < 16; ++i) {
        float* tile = wbuf + i * 256;

        // Step A: U = T * H.  A chunks = T[:,4kc:4kc+4] (A layout, b64 loads),
        //                     B chunks = hc (constant).
        const int abase = m * 16 + 2 * half;
        v2f a0 = *(const v2f*)(tile + abase + 0);
        v2f a1 = *(const v2f*)(tile + abase + 4);
        v2f a2 = *(const v2f*)(tile + abase + 8);
        v2f a3 = *(const v2f*)(tile + abase + 12);
        v8f u = {};
        u = __builtin_amdgcn_wmma_f32_16x16x4_f32(false, a0, false, hc[0], (short)0, u, false, false);
        u = __builtin_amdgcn_wmma_f32_16x16x4_f32(false, a1, false, hc[1], (short)0, u, false, false);
        u = __builtin_amdgcn_wmma_f32_16x16x4_f32(false, a2, false, hc[2], (short)0, u, false, false);
        u = __builtin_amdgcn_wmma_f32_16x16x4_f32(false, a3, false, hc[3], (short)0, u, false, false);

        // C/D layout -> row-major LDS (VGPR r = rows r / r+8 over lane halves)
#pragma unroll
        for (int r = 0; r < 8; ++r)
            tile[(r + 8 * half) * 16 + m] = u[r];

        // Step B: V = H * U.  A chunks = hc, B chunks = U rows (B layout).
        v2f b0, b1, b2, b3;
        const int rb = 2 * half;
        b0.x = tile[(rb + 0)  * 16 + m];  b0.y = tile[(rb + 1)  * 16 + m];
        b1.x = tile[(rb + 4)  * 16 + m];  b1.y = tile[(rb + 5)  * 16 + m];
        b2.x = tile[(rb + 8)  * 16 + m];  b2.y = tile[(rb + 9)  * 16 + m];
        b3.x = tile[(rb + 12) * 16 + m];  b3.y = tile[(rb + 13) * 16 + m];
        v8f v = {};
        v = __builtin_amdgcn_wmma_f32_16x16x4_f32(false, hc[0], false, b0, (short)0, v, false, false);
        v = __builtin_amdgcn_wmma_f32_16x16x4_f32(false, hc[1], false, b1, (short)0, v, false, false);
        v = __builtin_amdgcn_wmma_f32_16x16x4_f32(false, hc[2], false, b2, (short)0, v, false, false);
        v = __builtin_amdgcn_wmma_f32_16x16x4_f32(false, hc[3], false, b3, (short)0, v, false, false);

#pragma unroll
        for (int r = 0; r < 8; ++r)
            tile[(r + 8 * half) * 16 + m] = v[r];
    }

    // ---- Stage 3: H16 along i (register butterflies), scale, stream out ----
    const float scale = 1.0f / 64.0f;   // 1/sqrt(4096)
#pragma unroll
    for (int g = 0; g < 2; ++g) {
        const int p0 = lane * 8 + g * 4;        // (j,k) positions owned by lane
        v4f vals[16];
#pragma unroll
        for (int i = 0; i < 16; ++i)
            vals[i] = *(const v4f*)(wbuf + i * 256 + p0);
#pragma unroll
        for (int h = 1; h < 16; h <<= 1) {
#pragma unroll
            for (int a = 0; a < 16; a += 2 * h) {
#pragma unroll
                for (int b = 0; b < h; ++b) {
                    v4f L = vals[a + b], R = vals[a + b + h];
                    vals[a + b]     = L + R;
                    vals[a + b + h] = L - R;
                }
            }
        }
#pragma unroll
        for (int i = 0; i < 16; ++i)
            __builtin_nontemporal_store(vals[i] * scale, (v4f*)(orow + i * 256 + p0));
    }
}

extern "C" void kernel_launch(void* const* d_in, const int* in_sizes, int n_in,
                              void* d_out, int out_size, void* d_ws, size_t ws_size,
                              hipStream_t stream) {
    const float* x     = (const float*)d_in[0];
    const float* signs = (const float*)d_in[1];
    float* out = (float*)d_out;

    const int rows = in_sizes[0] / ROW_D;            // 16384
    const int blocks = rows / WAVES_PER_BLOCK;       // 4096 blocks, 1 row per wave
    rht4096_kernel<<<blocks, 128, 0, stream>>>(x, signs, out);
}